// GroupedAdditiveRouter_4183298146499
// MI455X (gfx1250) — compile-verified
//
#include <hip/hip_runtime.h>
#include <cstdint>
#include <cstddef>

// ---------------- problem constants (match reference) ----------------
#define NROWS   65536
#define DIMA    1024
#define DIMB    512
#define DIMC    256
#define NEXP    64      // E
#define TOPK    8

#define WAVES_PER_BLOCK 4
#define THREADS (WAVES_PER_BLOCK * 32)
#define ROWS_PER_BLOCK (WAVES_PER_BLOCK * 16)

typedef float v2f __attribute__((ext_vector_type(2)));
typedef float v8f __attribute__((ext_vector_type(8)));

static __device__ __forceinline__ v8f zero8() {
    v8f z = {0.f, 0.f, 0.f, 0.f, 0.f, 0.f, 0.f, 0.f};
    return z;
}

// D = A(16x4 f32) * B(4x16 f32) + C(16x16 f32), full fp32 precision.
static __device__ __forceinline__ v8f wmma_f32x4(v2f a, v2f b, v8f c) {
    return __builtin_amdgcn_wmma_f32_16x16x4_f32(
        /*neg_a=*/false, a, /*neg_b=*/false, b,
        /*c_mod=*/(short)0, c, /*reuse_a=*/false, /*reuse_b=*/false);
}

// One 16xE tile of  g[rows 16] @ W[D x E]  accumulated into acc[4] (4 N-tiles of 16).
// A-fragment layout (ISA 7.12.2, 32-bit A 16x4): lanes 0-15 hold K=kk,kk+1 of row m;
// lanes 16-31 hold K=kk+2,kk+3 of row m.  B mirrors it for the 4x16 tile.
template <int D>
static __device__ __forceinline__ void gemm_tile(const float* __restrict__ g,
                                                 const float* __restrict__ W,
                                                 int row0, int lane, v8f acc[4]) {
    const int m    = lane & 15;
    const int half = lane >> 4;
    const float* __restrict__ rp = g + (size_t)(row0 + m) * D;
    for (int kk = 0; kk < D; kk += 4) {
        if ((kk & 63) == 0)
            __builtin_prefetch(rp + kk + 256, 0, 1);   // global_prefetch_b8, 1KB ahead
        const int kb = kk + 2 * half;
        v2f a = *(const v2f*)(rp + kb);                // global_load_b64
#pragma unroll
        for (int t = 0; t < 4; ++t) {
            v2f b;
            b.x = W[(size_t)kb * NEXP + t * 16 + m];
            b.y = W[(size_t)(kb + 1) * NEXP + t * 16 + m];
            acc[t] = wmma_f32x4(a, b, acc[t]);
        }
    }
}

// C/D fragment -> row-major 16x64 tile in LDS (per-wave buffer).
static __device__ __forceinline__ void tile_to_lds(const v8f acc[4], float* __restrict__ buf,
                                                   int lane) {
    const int m    = lane & 15;
    const int half = lane >> 4;
#pragma unroll
    for (int t = 0; t < 4; ++t)
#pragma unroll
        for (int r = 0; r < 8; ++r)
            buf[(r + 8 * half) * NEXP + t * 16 + m] = acc[t][r];
}

// Coalesced store of the 16x64 LDS tile to a row-major [B,64] output.
static __device__ __forceinline__ void lds_tile_to_global(const float* __restrict__ buf,
                                                          float* __restrict__ dst,
                                                          int row0, int lane) {
    asm volatile("s_wait_dscnt 0" ::: "memory");       // per-wave LDS RAW fence
    const int m    = lane & 15;
    const int half = lane >> 4;
#pragma unroll
    for (int s = 0; s < 8; ++s) {
        const int r = s * 2 + half;
        const int c = m * 4;
        float4 v = *(const float4*)(buf + r * NEXP + c);       // ds_load_b128
        *(float4*)(dst + (size_t)(row0 + r) * NEXP + c) = v;   // global_store_b128
    }
}

__global__ __launch_bounds__(THREADS) void grouped_additive_router_kernel(
    const float* __restrict__ g_a, const float* __restrict__ g_b, const float* __restrict__ g_c,
    const float* __restrict__ W_a, const float* __restrict__ b_a,
    const float* __restrict__ W_b, const float* __restrict__ b_b,
    const float* __restrict__ W_c, const float* __restrict__ b_c,
    const float* __restrict__ bias, float* __restrict__ out) {
    __shared__ float buf0[WAVES_PER_BLOCK][16 * NEXP];   // pristine tile
    __shared__ float buf1[WAVES_PER_BLOCK][16 * NEXP];   // scratch / destructible

    const int tid  = threadIdx.x;
    const int w    = tid >> 5;
    const int lane = tid & 31;
    const int m    = lane & 15;
    const int row0 = blockIdx.x * ROWS_PER_BLOCK + w * 16;

    float* __restrict__ logits_out  = out;
    float* __restrict__ weights_out = out + 1 * (size_t)NROWS * NEXP;
    float* __restrict__ mask_out    = out + 2 * (size_t)NROWS * NEXP;
    float* __restrict__ ca_out      = out + 3 * (size_t)NROWS * NEXP;
    float* __restrict__ cb_out      = out + 4 * (size_t)NROWS * NEXP;
    float* __restrict__ cc_out      = out + 5 * (size_t)NROWS * NEXP;

    // per-lane column biases for the 4 N-tiles this lane touches
    float bA[4], bB[4], bC[4], bG[4];
#pragma unroll
    for (int t = 0; t < 4; ++t) {
        const int c = t * 16 + m;
        bA[t] = b_a[c]; bB[t] = b_b[c]; bC[t] = b_c[c]; bG[t] = bias[c];
    }

    v8f lacc[4];

    // ---- group A ----
    {
        v8f acc[4] = {zero8(), zero8(), zero8(), zero8()};
        gemm_tile<DIMA>(g_a, W_a, row0, lane, acc);
#pragma unroll
        for (int t = 0; t < 4; ++t) acc[t] = acc[t] + bA[t];
        tile_to_lds(acc, buf0[w], lane);
        lds_tile_to_global(buf0[w], ca_out, row0, lane);
#pragma unroll
        for (int t = 0; t < 4; ++t) lacc[t] = acc[t];
    }
    // ---- group B ----
    {
        v8f acc[4] = {zero8(), zero8(), zero8(), zero8()};
        gemm_tile<DIMB>(g_b, W_b, row0, lane, acc);
#pragma unroll
        for (int t = 0; t < 4; ++t) acc[t] = acc[t] + bB[t];
        tile_to_lds(acc, buf0[w], lane);
        lds_tile_to_global(buf0[w], cb_out, row0, lane);
#pragma unroll
        for (int t = 0; t < 4; ++t) lacc[t] = lacc[t] + acc[t];
    }
    // ---- group C ----
    {
        v8f acc[4] = {zero8(), zero8(), zero8(), zero8()};
        gemm_tile<DIMC>(g_c, W_c, row0, lane, acc);
#pragma unroll
        for (int t = 0; t < 4; ++t) acc[t] = acc[t] + bC[t];
        tile_to_lds(acc, buf0[w], lane);
        lds_tile_to_global(buf0[w], cc_out, row0, lane);
#pragma unroll
        for (int t = 0; t < 4; ++t) lacc[t] = lacc[t] + acc[t] + bG[t];
    }

    // ---- logits ----
    tile_to_lds(lacc, buf0[w], lane);
    lds_tile_to_global(buf0[w], logits_out, row0, lane);

    // copy pristine logits -> destructible buffer (b128 both ways)
#pragma unroll
    for (int i = 0; i < 8; ++i) {
        const int idx = (i * 32 + lane) * 4;
        *(float4*)(buf1[w] + idx) = *(const float4*)(buf0[w] + idx);
    }
    asm volatile("s_wait_dscnt 0" ::: "memory");

    // ---- top-8 selection + masked softmax: one row per lane (lanes 0..15) ----
    unsigned long long selmask = 0ULL;
    float maxsel = 0.f, invS = 0.f;
    if (lane < 16) {
        float* __restrict__ rowv       = buf1[w] + lane * NEXP;  // destroyed
        const float* __restrict__ rowl = buf0[w] + lane * NEXP;  // pristine
#pragma unroll 1
        for (int p = 0; p < TOPK; ++p) {
            float best = -__builtin_inff();
            int bestj = 0;
            for (int j = 0; j < NEXP; ++j) {
                const float v = rowv[j];
                if (v > best) { best = v; bestj = j; }   // strict '>' => first index on ties
            }
            rowv[bestj] = -__builtin_inff();
            selmask |= (1ULL << bestj);
            if (p == 0) maxsel = best;
        }
        float S = 0.f;
        for (int j = 0; j < NEXP; ++j)
            if ((selmask >> j) & 1ULL) S += __expf(rowl[j] - maxsel);
        invS = 1.0f / S;
        for (int j = 0; j < NEXP; ++j) {
            const float wgt = ((selmask >> j) & 1ULL) ? __expf(rowl[j] - maxsel) * invS : 0.0f;
            rowv[j] = wgt;
        }
    }
    lds_tile_to_global(buf1[w], weights_out, row0, lane);

    if (lane < 16) {
        float* __restrict__ rowv = buf1[w] + lane * NEXP;
        for (int j = 0; j < NEXP; ++j)
            rowv[j] = ((selmask >> j) & 1ULL) ? 1.0f : 0.0f;
    }
    lds_tile_to_global(buf1[w], mask_out, row0, lane);
}

extern "C" void kernel_launch(void* const* d_in, const int* in_sizes, int n_in,
                              void* d_out, int out_size, void* d_ws, size_t ws_size,
                              hipStream_t stream) {
    (void)in_sizes; (void)n_in; (void)out_size; (void)d_ws; (void)ws_size;
    const float* g_a  = (const float*)d_in[0];
    const float* g_b  = (const float*)d_in[1];
    const float* g_c  = (const float*)d_in[2];
    const float* W_a  = (const float*)d_in[3];
    const float* b_a  = (const float*)d_in[4];
    const float* W_b  = (const float*)d_in[5];
    const float* b_b  = (const float*)d_in[6];
    const float* W_c  = (const float*)d_in[7];
    const float* b_c  = (const float*)d_in[8];
    const float* bias = (const float*)d_in[9];
    // d_in[10] = top_k (== 8, compile-time constant here)
    float* out = (float*)d_out;

    dim3 grid(NROWS / ROWS_PER_BLOCK);   // 1024 blocks
    dim3 block(THREADS);                 // 128 threads = 4 waves
    grouped_additive_router_kernel<<<grid, block, 0, stream>>>(
        g_a, g_b, g_c, W_a, b_a, W_b, b_b, W_c, b_c, bias, out);
}